// GraphDecoder_49134425866405
// MI455X (gfx1250) — compile-verified
//
#include <hip/hip_runtime.h>
#include <cstddef>

// ---------------------------------------------------------------------------
// GATConv x3 for MI455X (gfx1250, wave32).
// GEMMs use V_WMMA_F32_16X16X4_F32 with W staged in LDS in a K-paired
// layout so every B fragment is one aligned ds_load_b64 (even VGPR pair).
// Edge phase: wave-per-edge gather/scatter through L2 (f32 atomics),
// segment-max softmax via monotone ordered-uint atomicMax.
// ---------------------------------------------------------------------------

#define GAT_N     50000
#define GAT_E     800000
#define GAT_HID   128
#define GAT_OUT   64
#define NEG_SLOPE 0.2f

typedef float v2f __attribute__((ext_vector_type(2)));
typedef float v8f __attribute__((ext_vector_type(8)));

// ----- monotone float<->uint mapping for atomic max on floats ---------------
__device__ __forceinline__ unsigned f2ord(float f) {
  unsigned u = __float_as_uint(f);
  return (u & 0x80000000u) ? ~u : (u | 0x80000000u);
}
__device__ __forceinline__ float ord2f(unsigned u) {
  return __uint_as_float((u & 0x80000000u) ? (u & 0x7FFFFFFFu) : ~u);
}
#define ORD_NEG_INF 0x007FFFFFu   // f2ord(-inf)

// ---------------------------------------------------------------------------
// fill kernel (used for zeroing accumulators and initializing segment-max)
// ---------------------------------------------------------------------------
__global__ void fill_u32_kernel(unsigned* __restrict__ p, unsigned val, long long n) {
  long long i = (long long)blockIdx.x * blockDim.x + threadIdx.x;
  if (i < n) p[i] = val;
}

// ---------------------------------------------------------------------------
// GEMM: XL[N, NCOLS] = X[N,128] @ W[128, NCOLS]   via v_wmma_f32_16x16x4_f32
// 256 threads = 8 waves; each wave owns one 16-row tile across all NCOLS.
//
// LDS holds W in K-paired layout:
//   Wl[kp * NCOLS + n] = { W[2*kp][n], W[2*kp+1][n] }      (kp = 0..63)
// so lane's B fragment (K = k+khalf, k+khalf+1 at column n) is a single
// 8-byte aligned LDS load into an even VGPR pair.
// ---------------------------------------------------------------------------
template <int NCOLS>
__global__ __launch_bounds__(256) void gat_gemm_wmma(
    const float* __restrict__ X, const float* __restrict__ W,
    float* __restrict__ XL, int nrows) {
  __shared__ v2f Wl[64 * NCOLS];     // 128*NCOLS floats (64KB or 32KB)
  float* Wlf = (float*)Wl;

  const int tid = threadIdx.x;
  // cooperative load of W (row-major [128][NCOLS]) into K-paired LDS layout
  for (int i = tid * 4; i < 128 * NCOLS; i += 256 * 4) {
    const float4 wv = *(const float4*)(&W[i]);
    const int k = i / NCOLS;
    const int n = i % NCOLS;                       // NCOLS % 4 == 0
    const int base = (k >> 1) * 2 * NCOLS + (k & 1) + 2 * n;
    Wlf[base + 0] = wv.x;
    Wlf[base + 2] = wv.y;
    Wlf[base + 4] = wv.z;
    Wlf[base + 6] = wv.w;
  }
  __syncthreads();

  const int wave  = tid >> 5;
  const int lane  = tid & 31;
  const int tile  = blockIdx.x * 8 + wave;   // 16-row tile index
  if (tile * 16 >= nrows) return;            // N % 16 == 0, tiles are full

  const int row0  = tile * 16;
  const int lrow  = lane & 15;               // A: M index; B/C: N index
  const int kh    = lane >> 4;               // upper half-wave holds K+2,K+3

  const float* xrow = X + (size_t)(row0 + lrow) * 128;
  const v2f* Wrow   = Wl + (size_t)kh * NCOLS + lrow;  // + kp*NCOLS per step

  v8f acc[NCOLS / 16];
#pragma unroll
  for (int nt = 0; nt < NCOLS / 16; ++nt) acc[nt] = {};

  for (int k = 0; k < 128; k += 4) {
    // A fragment: 16x4 f32; lane<16 -> K=k..k+1, lane>=16 -> K=k+2..k+3
    v2f a = *(const v2f*)(xrow + k + kh * 2);
    const v2f* wp = Wrow + (size_t)(k >> 1) * NCOLS;
#pragma unroll
    for (int nt = 0; nt < NCOLS / 16; ++nt) {
      v2f b = wp[nt * 16];                   // single ds_load_b64
      acc[nt] = __builtin_amdgcn_wmma_f32_16x16x4_f32(
          /*neg_a=*/false, a, /*neg_b=*/false, b,
          /*c_mod=*/(short)0, acc[nt], /*reuse_a=*/false, /*reuse_b=*/false);
    }
  }

  // C/D layout: VGPR r, lanes 0-15 -> M=r; lanes 16-31 -> M=8+r; N=lane&15
  const int moff = kh * 8;
#pragma unroll
  for (int nt = 0; nt < NCOLS / 16; ++nt) {
#pragma unroll
    for (int r = 0; r < 8; ++r) {
      XL[(size_t)(row0 + moff + r) * NCOLS + nt * 16 + lrow] = acc[nt][r];
    }
  }
}

// ---------------------------------------------------------------------------
// attention coefficients: a_s[n,h] = <xl[n,h,:], att_src[h,:]>, same for a_d
// ---------------------------------------------------------------------------
template <int H, int C>
__global__ __launch_bounds__(256) void gat_att_kernel(
    const float* __restrict__ XL, const float* __restrict__ att_s,
    const float* __restrict__ att_d, float* __restrict__ as_out,
    float* __restrict__ ad_out, int nrows) {
  int i = blockIdx.x * blockDim.x + threadIdx.x;   // i = row*H + h
  if (i >= nrows * H) return;
  const int row = i / H, h = i % H;
  const float* v = XL + (size_t)row * (H * C) + h * C;
  float s = 0.f, d = 0.f;
#pragma unroll 8
  for (int c = 0; c < C; ++c) {
    const float xv = v[c];
    s += xv * att_s[h * C + c];
    d += xv * att_d[h * C + c];
  }
  as_out[i] = s;
  ad_out[i] = d;
}

// ---------------------------------------------------------------------------
// pass 1 over edges: segment max of leaky-relu logits (ordered-uint atomicMax)
// ---------------------------------------------------------------------------
template <int H>
__global__ __launch_bounds__(256) void edge_max_kernel(
    const int* __restrict__ srcs, const int* __restrict__ dsts,
    int nedges, int nnodes, const float* __restrict__ as_,
    const float* __restrict__ ad_, unsigned* __restrict__ m_ord) {
  int i = blockIdx.x * blockDim.x + threadIdx.x;   // i = e*H + h
  if (i >= (nedges + nnodes) * H) return;
  const int e = i / H, h = i % H;
  int s, d;
  if (e < nedges) { s = srcs[e]; d = dsts[e]; } else { s = d = e - nedges; }
  float t = as_[s * H + h] + ad_[d * H + h];
  t = t > 0.f ? t : t * NEG_SLOPE;
  atomicMax(&m_ord[d * H + h], f2ord(t));
}

// ---------------------------------------------------------------------------
// pass 2 over edges: denom[dst,h] += exp(t - m[dst,h])
// ---------------------------------------------------------------------------
template <int H>
__global__ __launch_bounds__(256) void edge_expsum_kernel(
    const int* __restrict__ srcs, const int* __restrict__ dsts,
    int nedges, int nnodes, const float* __restrict__ as_,
    const float* __restrict__ ad_, const unsigned* __restrict__ m_ord,
    float* __restrict__ denom) {
  int i = blockIdx.x * blockDim.x + threadIdx.x;
  if (i >= (nedges + nnodes) * H) return;
  const int e = i / H, h = i % H;
  int s, d;
  if (e < nedges) { s = srcs[e]; d = dsts[e]; } else { s = d = e - nedges; }
  float t = as_[s * H + h] + ad_[d * H + h];
  t = t > 0.f ? t : t * NEG_SLOPE;
  const float m = ord2f(m_ord[d * H + h]);
  atomicAdd(&denom[d * H + h], __expf(t - m));
}

// ---------------------------------------------------------------------------
// pass 3 over edges: out[dst] += alpha * xl[src]     (wave32 per edge)
// lane owns HC/32 contiguous channels -> one float4 (or float2) gather and
// per-channel f32 L2 atomics on the scatter side.
// ---------------------------------------------------------------------------
template <int H, int C>
__global__ __launch_bounds__(256) void edge_accum_kernel(
    const int* __restrict__ srcs, const int* __restrict__ dsts,
    int nedges, int nnodes, const float* __restrict__ as_,
    const float* __restrict__ ad_, const unsigned* __restrict__ m_ord,
    const float* __restrict__ denom, const float* __restrict__ XL,
    float* __restrict__ OUT) {
  constexpr int HC  = H * C;
  constexpr int PER = HC / 32;                 // channels per lane (4 or 2)
  const int wid  = (int)((blockIdx.x * blockDim.x + threadIdx.x) >> 5);
  const int lane = threadIdx.x & 31;
  if (wid >= nedges + nnodes) return;
  int s, d;
  if (wid < nedges) { s = srcs[wid]; d = dsts[wid]; } else { s = d = wid - nedges; }

  const int c0 = lane * PER;
  const int h  = c0 / C;

  const float* xs = XL + (size_t)s * HC + c0;
  __builtin_prefetch(xs, 0, 0);               // global_prefetch_b8

  float t = as_[s * H + h] + ad_[d * H + h];
  t = t > 0.f ? t : t * NEG_SLOPE;
  const float m     = ord2f(m_ord[d * H + h]);
  const float alpha = __expf(t - m) / (denom[d * H + h] + 1e-16f);

  float* od = OUT + (size_t)d * HC + c0;
#pragma unroll
  for (int j = 0; j < PER; ++j) atomicAdd(&od[j], xs[j] * alpha);
}

// ---------------------------------------------------------------------------
// epilogue: buf[n, hc] += bias[hc]; optional ReLU
// ---------------------------------------------------------------------------
__global__ __launch_bounds__(256) void bias_act_kernel(
    float* __restrict__ buf, const float* __restrict__ bias, int hc,
    long long total, int relu) {
  long long i = (long long)blockIdx.x * blockDim.x + threadIdx.x;
  if (i >= total) return;
  float v = buf[i] + bias[(int)(i % hc)];
  if (relu) v = v > 0.f ? v : 0.f;
  buf[i] = v;
}

// ---------------------------------------------------------------------------
// host-side: one full GAT layer
// ---------------------------------------------------------------------------
static inline unsigned cdiv(long long a, long long b) {
  return (unsigned)((a + b - 1) / b);
}

template <int H, int C>
static void run_gat_layer(const float* feat_in, const float* W,
                          const float* att_s, const float* att_d,
                          const float* bias, const int* src, const int* dst,
                          float* xl, float* accum, float* as_, float* ad_,
                          unsigned* m_ord, float* denom, int relu,
                          hipStream_t stream) {
  constexpr int HC = H * C;
  const int n = GAT_N, ne = GAT_E;
  const long long nEH = (long long)(ne + n) * H;

  // xl = feat_in @ W  (WMMA)
  gat_gemm_wmma<HC><<<cdiv(n / 16, 8), 256, 0, stream>>>(feat_in, W, xl, n);
  // attention logits
  gat_att_kernel<H, C><<<cdiv((long long)n * H, 256), 256, 0, stream>>>(
      xl, att_s, att_d, as_, ad_, n);
  // init segment-max / denom / accumulator
  fill_u32_kernel<<<cdiv((long long)n * H, 256), 256, 0, stream>>>(
      m_ord, ORD_NEG_INF, (long long)n * H);
  fill_u32_kernel<<<cdiv((long long)n * H, 256), 256, 0, stream>>>(
      (unsigned*)denom, 0u, (long long)n * H);
  fill_u32_kernel<<<cdiv((long long)n * HC, 256), 256, 0, stream>>>(
      (unsigned*)accum, 0u, (long long)n * HC);
  // softmax + aggregate
  edge_max_kernel<H><<<cdiv(nEH, 256), 256, 0, stream>>>(
      src, dst, ne, n, as_, ad_, m_ord);
  edge_expsum_kernel<H><<<cdiv(nEH, 256), 256, 0, stream>>>(
      src, dst, ne, n, as_, ad_, m_ord, denom);
  edge_accum_kernel<H, C><<<cdiv((long long)(ne + n) * 32, 256), 256, 0, stream>>>(
      src, dst, ne, n, as_, ad_, m_ord, denom, xl, accum);
  // bias (+ReLU)
  bias_act_kernel<<<cdiv((long long)n * HC, 256), 256, 0, stream>>>(
      accum, bias, HC, (long long)n * HC, relu);
}

// ---------------------------------------------------------------------------
extern "C" void kernel_launch(void* const* d_in, const int* in_sizes, int n_in,
                              void* d_out, int out_size, void* d_ws,
                              size_t ws_size, hipStream_t stream) {
  (void)in_sizes; (void)n_in; (void)out_size; (void)ws_size;

  const float* x   = (const float*)d_in[0];
  const int*   ei  = (const int*)d_in[1];      // [3][2][E]
  const float* W1  = (const float*)d_in[2];
  const float* as1 = (const float*)d_in[3];
  const float* ad1 = (const float*)d_in[4];
  const float* b1  = (const float*)d_in[5];
  const float* W2  = (const float*)d_in[6];
  const float* as2 = (const float*)d_in[7];
  const float* ad2 = (const float*)d_in[8];
  const float* b2  = (const float*)d_in[9];
  const float* W3  = (const float*)d_in[10];
  const float* as3 = (const float*)d_in[11];
  const float* ad3 = (const float*)d_in[12];
  const float* b3  = (const float*)d_in[13];
  float* out = (float*)d_out;

  // workspace layout (floats)
  float* ws = (float*)d_ws;
  const size_t NF = (size_t)GAT_N * GAT_HID;
  float*    A     = ws;                         // xl scratch [N,128]
  float*    B     = ws + NF;                    // accum / feature [N,128]
  float*    as_   = B + NF;                     // [N,4]
  float*    ad_   = as_ + (size_t)GAT_N * 4;    // [N,4]
  unsigned* m_ord = (unsigned*)(ad_ + (size_t)GAT_N * 4);
  float*    denom = (float*)(m_ord + (size_t)GAT_N * 4);

  const int* src0 = ei + 0 * 2 * GAT_E;
  const int* dst0 = src0 + GAT_E;
  const int* src1 = ei + 1 * 2 * GAT_E;
  const int* dst1 = src1 + GAT_E;
  const int* src2 = ei + 2 * 2 * GAT_E;
  const int* dst2 = src2 + GAT_E;

  // layer 1: x -> B (ReLU)
  run_gat_layer<4, 32>(x, W1, as1, ad1, b1, src0, dst0,
                       A, B, as_, ad_, m_ord, denom, /*relu=*/1, stream);
  // layer 2: B -> B (gemm reads B into A first, then B is recycled as accum)
  run_gat_layer<4, 32>(B, W2, as2, ad2, b2, src1, dst1,
                       A, B, as_, ad_, m_ord, denom, /*relu=*/1, stream);
  // layer 3: B -> d_out (no ReLU), single head, 64 channels
  run_gat_layer<1, 64>(B, W3, as3, ad3, b3, src2, dst2,
                       A, out, as_, ad_, m_ord, denom, /*relu=*/0, stream);
}